// VectorQuantizerClassic_36799279792262
// MI455X (gfx1250) — compile-verified
//
#include <hip/hip_runtime.h>

typedef __attribute__((ext_vector_type(2))) float v2f;
typedef __attribute__((ext_vector_type(8))) float v8f;

#define N_ROWS   131072     // 32 * 64 * 64
#define N_CODES  2048
#define E_DIM    8
#define HW       4096       // 64*64
#define TEMP_INV 10.0f      // 1 / TEMPERATURE
#define KX       (-14.4269504088896340f)   // -TEMP_INV * log2(e)
#define LN2      0.69314718055994531f

// workspace (floats): [0]=sum of d_min over rows, [1]=sum of (sum_j p*logp) over rows,
//                     [2 .. 2+N_CODES) = sum of probs per code over rows
__global__ void vq_init_ws(float* __restrict__ ws) {
    int i = blockIdx.x * blockDim.x + threadIdx.x;
    if (i < 2 + N_CODES) ws[i] = 0.0f;
}

__global__ __launch_bounds__(256) void vq_main(const float* __restrict__ z,
                                               const float* __restrict__ emb,
                                               float* __restrict__ zq_out,
                                               float* __restrict__ idx_out,
                                               float* __restrict__ ws) {
    __shared__ float ldsEmb[N_CODES * E_DIM];   // 64 KB codebook
    __shared__ float ldsEnorm[N_CODES];         // |e_j|^2
    __shared__ float ldsAvgP[N_CODES];          // block-partial sum of probs
    __shared__ float ldsScal[2];                // block-partial (sum dmin, sum p*logp)

    const int tid = threadIdx.x;

    // Cooperative load of the codebook into LDS
    {
        const float4* g4 = (const float4*)emb;
        float4* l4 = (float4*)ldsEmb;
        for (int i = tid; i < (N_CODES * E_DIM) / 4; i += 256) l4[i] = g4[i];
    }
    for (int i = tid; i < N_CODES; i += 256) ldsAvgP[i] = 0.0f;
    if (tid < 2) ldsScal[tid] = 0.0f;
    __syncthreads();
    for (int i = tid; i < N_CODES; i += 256) {
        float sq = 0.0f;
#pragma unroll
        for (int c = 0; c < E_DIM; ++c) { float v = ldsEmb[i * E_DIM + c]; sq += v * v; }
        ldsEnorm[i] = sq;
    }
    __syncthreads();

    const int wave = tid >> 5;
    const int lane = tid & 31;
    const int half = lane >> 4;   // which 16-lane half
    const int l15  = lane & 15;
    const int kb   = half * 2;    // K base (f32 A/B layout: VGPR0 = K0|K2, VGPR1 = K1|K3)
    const int n0   = blockIdx.x * 128 + wave * 16;

    // ---- A matrix (16 rows x 8 dims) resident in registers, two K-chunks ----
    const int row = n0 + l15;
    const int bb  = row >> 12;
    const int pp  = row & (HW - 1);
    const float* zr = z + (size_t)bb * E_DIM * HW + pp;
    v2f a0, a1;
    a0.x = zr[(kb + 0) * HW]; a0.y = zr[(kb + 1) * HW];
    a1.x = zr[(kb + 4) * HW]; a1.y = zr[(kb + 5) * HW];

    // |z_row|^2 : each half covers 4 disjoint dims
    float znPart = a0.x * a0.x + a0.y * a0.y + a1.x * a1.x + a1.y * a1.y;
    float zn = znPart + __shfl_xor(znPart, 16, 32);
    float znr[8];
#pragma unroll
    for (int r = 0; r < 8; ++r) znr[r] = __shfl(zn, r + 8 * half, 32);

    // B-tile fetch helper (operands for one 16-code tile)
    #define LOAD_B(J0, B0, B1, EN) do {                         \
        const int _c = (J0) + l15;                              \
        const float* _eb = &ldsEmb[_c * E_DIM + kb];            \
        (B0).x = _eb[0]; (B0).y = _eb[1];                       \
        (B1).x = _eb[4]; (B1).y = _eb[5];                       \
        (EN) = ldsEnorm[_c];                                    \
    } while (0)

    // ============ Pass A: argmin over all codes (no transcendentals) ============
    float dmn[8];
    int   jmn[8];
#pragma unroll
    for (int r = 0; r < 8; ++r) { dmn[r] = 1e30f; jmn[r] = 0; }

    {
        v2f b0, b1; float en;
        LOAD_B(0, b0, b1, en);
        for (int j0 = 0; j0 < N_CODES; j0 += 16) {
            v2f nb0, nb1; float nen;
            LOAD_B((j0 + 16) & (N_CODES - 1), nb0, nb1, nen);   // branch-free prefetch
            v8f acc = {};
            acc = __builtin_amdgcn_wmma_f32_16x16x4_f32(false, a0, false, b0, (short)0, acc, false, false);
            acc = __builtin_amdgcn_wmma_f32_16x16x4_f32(false, a1, false, b1, (short)0, acc, false, false);
            const int code = j0 + l15;
#pragma unroll
            for (int r = 0; r < 8; ++r) {
                float d = (znr[r] + en) - 2.0f * acc[r];
                if (d < dmn[r]) { dmn[r] = d; jmn[r] = code; }
            }
            b0 = nb0; b1 = nb1; en = nen;
        }
    }
    // reduce (dmin, argmin) across the 16 lanes of each half (first-index tie-break)
#pragma unroll
    for (int off = 1; off <= 8; off <<= 1) {
#pragma unroll
        for (int r = 0; r < 8; ++r) {
            float dmo = __shfl_xor(dmn[r], off, 32);
            int   jo  = __shfl_xor(jmn[r], off, 32);
            if (dmo < dmn[r] || (dmo == dmn[r] && jo < jmn[r])) { dmn[r] = dmo; jmn[r] = jo; }
        }
    }

    // exact row max of logits: m = -10 * dmin
    float m[8];
#pragma unroll
    for (int r = 0; r < 8; ++r) m[r] = -TEMP_INV * dmn[r];

    // per-row outputs: indices + z_q gather (STE value == embedding[argmin])
    if (l15 < 8) {
        const int r = l15;
        const int orow = n0 + r + 8 * half;
        const int ob = orow >> 12, op = orow & (HW - 1);
        const int j = jmn[r];
        idx_out[orow] = (float)j;
#pragma unroll
        for (int c = 0; c < E_DIM; ++c)
            zq_out[(ob * E_DIM + c) * HW + op] = ldsEmb[j * E_DIM + c];
    }

    // ============ Pass B: s = sum exp(lg-m), t2 = sum exp(lg-m)*(d-dmin) ============
    float s[8], t2[8], c1[8];
#pragma unroll
    for (int r = 0; r < 8; ++r) { s[r] = 0.0f; t2[r] = 0.0f; c1[r] = znr[r] - dmn[r]; }

    {
        v2f b0, b1; float en;
        LOAD_B(0, b0, b1, en);
        for (int j0 = 0; j0 < N_CODES; j0 += 16) {
            v2f nb0, nb1; float nen;
            LOAD_B((j0 + 16) & (N_CODES - 1), nb0, nb1, nen);
            v8f acc = {};
            acc = __builtin_amdgcn_wmma_f32_16x16x4_f32(false, a0, false, b0, (short)0, acc, false, false);
            acc = __builtin_amdgcn_wmma_f32_16x16x4_f32(false, a1, false, b1, (short)0, acc, false, false);
#pragma unroll
            for (int r = 0; r < 8; ++r) {
                float x  = (c1[r] + en) - 2.0f * acc[r];        // d - dmin  (>= 0)
                float ee = __builtin_amdgcn_exp2f(KX * x);      // exp(lg - m), 1 mul + v_exp
                s[r]  += ee;
                t2[r] += ee * x;
            }
            b0 = nb0; b1 = nb1; en = nen;
        }
    }
    // plain additive reduction across the 16 lanes of each half
#pragma unroll
    for (int off = 1; off <= 8; off <<= 1) {
#pragma unroll
        for (int r = 0; r < 8; ++r) {
            s[r]  += __shfl_xor(s[r],  off, 32);
            t2[r] += __shfl_xor(t2[r], off, 32);
        }
    }

    float logZ[8];
#pragma unroll
    for (int r = 0; r < 8; ++r) logZ[r] = m[r] + __builtin_amdgcn_logf(s[r]) * LN2;

    // scalar partials: dmin IS sum_c (z_q - z)^2 ;
    // sum_j p*logp = t/s - logZ with t = -10*t2 + m*s
    if (l15 == 0) {
        float sd = 0.0f, sp = 0.0f;
#pragma unroll
        for (int r = 0; r < 8; ++r) {
            sd += dmn[r];
            float t = -TEMP_INV * t2[r] + m[r] * s[r];
            sp += t / s[r] - logZ[r];
        }
        atomicAdd(&ldsScal[0], sd);
        atomicAdd(&ldsScal[1], sp);
    }

    // ============ Pass C: probs accumulation (logZ folded into exp arg) ============
    float c3[8];
#pragma unroll
    for (int r = 0; r < 8; ++r) c3[r] = znr[r] + logZ[r] * (1.0f / TEMP_INV);

    {
        v2f b0, b1; float en;
        LOAD_B(0, b0, b1, en);
        for (int j0 = 0; j0 < N_CODES; j0 += 16) {
            v2f nb0, nb1; float nen;
            LOAD_B((j0 + 16) & (N_CODES - 1), nb0, nb1, nen);
            v8f acc = {};
            acc = __builtin_amdgcn_wmma_f32_16x16x4_f32(false, a0, false, b0, (short)0, acc, false, false);
            acc = __builtin_amdgcn_wmma_f32_16x16x4_f32(false, a1, false, b1, (short)0, acc, false, false);
            const int code = j0 + l15;
            float psum = 0.0f;
#pragma unroll
            for (int r = 0; r < 8; ++r) {
                float x = (c3[r] + en) - 2.0f * acc[r];         // (logZ - lg)/10
                psum += __builtin_amdgcn_exp2f(KX * x);         // = prob
            }
            atomicAdd(&ldsAvgP[code], psum);   // ds_add_f32
            b0 = nb0; b1 = nb1; en = nen;
        }
    }
    __syncthreads();

    for (int i = tid; i < N_CODES; i += 256) atomicAdd(&ws[2 + i], ldsAvgP[i]);
    if (tid == 0) { atomicAdd(&ws[0], ldsScal[0]); atomicAdd(&ws[1], ldsScal[1]); }
}

__global__ __launch_bounds__(256) void vq_finalize(const float* __restrict__ ws,
                                                   float* __restrict__ out_scalars) {
    __shared__ float red[256];
    const int tid = threadIdx.x;
    float part = 0.0f;
    for (int i = tid; i < N_CODES; i += 256) {
        float ap = ws[2 + i] * (1.0f / (float)N_ROWS);
        part += ap * __logf(ap + 1e-5f);
    }
    red[tid] = part;
    __syncthreads();
    for (int st = 128; st > 0; st >>= 1) {
        if (tid < st) red[tid] += red[tid + st];
        __syncthreads();
    }
    if (tid == 0) {
        float avg_entropy    = -red[0];
        float sample_entropy = -ws[1] / (float)N_ROWS;
        float mean_sq        = ws[0] / (float)(N_ROWS * E_DIM);
        out_scalars[0] = mean_sq;                                   // vq_loss
        out_scalars[1] = 0.25f * mean_sq;                           // commit_loss (BETA)
        out_scalars[2] = 0.1f * (sample_entropy - avg_entropy);     // entropy_loss
    }
}

extern "C" void kernel_launch(void* const* d_in, const int* in_sizes, int n_in,
                              void* d_out, int out_size, void* d_ws, size_t ws_size,
                              hipStream_t stream) {
    const float* z   = (const float*)d_in[0];   // [32, 8, 64, 64] f32
    const float* emb = (const float*)d_in[1];   // [2048, 8] f32
    float* out = (float*)d_out;
    float* zq      = out;                        // 1048576 floats
    float* scalars = out + (N_ROWS * E_DIM);     // vq, commit, entropy
    float* idxf    = out + (N_ROWS * E_DIM) + 3; // 131072 indices (as float)
    float* ws = (float*)d_ws;

    vq_init_ws<<<(2 + N_CODES + 255) / 256, 256, 0, stream>>>(ws);
    vq_main<<<N_ROWS / 128, 256, 0, stream>>>(z, emb, zq, idxf, ws);
    vq_finalize<<<1, 256, 0, stream>>>(ws, scalars);
}